// InternalGraphConvolutionLayer_41051297415546
// MI455X (gfx1250) — compile-verified
//
#include <hip/hip_runtime.h>
#include <hip/hip_bf16.h>

typedef __attribute__((ext_vector_type(2))) float v2f;
typedef __attribute__((ext_vector_type(8))) float v8f;

#define DD 128   // node representation size

// ---------------------------------------------------------------------------
// Kernel A: agg[n,:] = sum_{e : segment_ids[e]==n} emb[neighbor_ids[e], :]
// segment_ids is sorted -> each block binary-searches its edge range.
// 4 waves/block: wave w handles edges start+w, start+w+4, ... and each lane
// loads float4 (global_load_b128, 512B per row per wave). 4 partial vectors
// are combined through LDS. This maximizes memory-level parallelism on the
// roofline-dominant gather (E*D*4B ~ 327MB, L2-resident embedding table).
// ---------------------------------------------------------------------------
__global__ void seg_aggregate(const int* __restrict__ neighbor_ids,
                              const int* __restrict__ segment_ids,
                              const float* __restrict__ emb,
                              float* __restrict__ agg,
                              int E) {
  const int node = blockIdx.x;
  const int wave = threadIdx.x >> 5;     // 0..3
  const int lane = threadIdx.x & 31;

  __shared__ int   s_range[2];
  __shared__ float s_part[4][DD];

  if (threadIdx.x < 2) {
    const int target = node + (int)threadIdx.x;   // lower_bound(target)
    int lo = 0, hi = E;
    while (lo < hi) {
      int mid = (lo + hi) >> 1;
      if (segment_ids[mid] < target) lo = mid + 1; else hi = mid;
    }
    s_range[threadIdx.x] = lo;
  }
  __syncthreads();
  const int start = s_range[0];
  const int end   = s_range[1];

  float4 acc = make_float4(0.f, 0.f, 0.f, 0.f);
  for (int e = start + wave; e < end; e += 4) {
    const int nb = neighbor_ids[e];
    const float4 v = *(const float4*)(emb + (size_t)nb * DD + lane * 4);
    acc.x += v.x; acc.y += v.y; acc.z += v.z; acc.w += v.w;
  }
  *(float4*)(&s_part[wave][lane * 4]) = acc;
  __syncthreads();

  const int d = threadIdx.x;
  agg[(size_t)node * DD + d] =
      s_part[0][d] + s_part[1][d] + s_part[2][d] + s_part[3][d];
}

// ---------------------------------------------------------------------------
// Zero the 128-float atomic accumulation buffer.
// ---------------------------------------------------------------------------
__global__ void zero_result(float* __restrict__ result) {
  result[threadIdx.x] = 0.0f;
}

// ---------------------------------------------------------------------------
// Kernel B: per_node = relu(emb[node_ids] @ W^T + agg @ M^T);
//           result[j] += column sums.  One wave handles 16 nodes x 128 cols
// with V_WMMA_F32_16X16X4_F32 (fp32, K=4), accumulating both GEMMs into the
// same C so ReLU applies to the combined pre-activation.
//
// Inner loop loads all 8 B tiles into distinct registers BEFORE issuing the
// 8 WMMAs so the loads clause together (s_clause + staggered s_wait_loadcnt)
// instead of a serialized load->wait0->wmma chain.
//
// Tail handling without per-iteration EXEC branches:
//   - whole-wave early exit when node_base >= N (wave-uniform)
//   - row index clamped for loads; invalid rows masked out in the colsum
//     (WMMA rows are independent, so garbage rows never touch valid ones).
//
// VGPR layouts (ISA 7.12.2, 32-bit):
//   A 16x4 : lane<16 holds {A[l][k0],A[l][k0+1]}, lane>=16 K offsets 2,3
//   B 4x16 : lane<16 holds {B[k0][l],B[k0+1][l]}, lane>=16 K offsets 2,3
//            with B[k][j] = W[j][k]  (we need X @ W^T)
//   C/D    : vgpr r = row (r + 8*(lane>=16)), col = lane & 15
// ---------------------------------------------------------------------------
__global__ void gemm_relu_colsum(const int* __restrict__ node_ids,
                                 const float* __restrict__ emb,
                                 const float* __restrict__ agg,
                                 const float* __restrict__ W,
                                 const float* __restrict__ M,
                                 float* __restrict__ result,
                                 int N) {
  const int wave   = threadIdx.x >> 5;
  const int lane   = threadIdx.x & 31;
  const int rowsel = lane & 15;
  const int half   = (lane >> 4) & 1;
  const int kshift = half * 2;             // K offset 0 or 2

  const int node_base = (blockIdx.x * (int)(blockDim.x >> 5) + wave) * 16;
  if (node_base >= N) return;              // wave-uniform exit

  const int row  = node_base + rowsel;
  const int rowc = (row < N) ? row : (N - 1);   // clamped: loads always legal
  const int nid  = node_ids[rowc];

  v8f acc[8];
  v8f zero = {};
  #pragma unroll
  for (int jt = 0; jt < 8; ++jt) acc[jt] = zero;

  // pass 0: A = emb[node_ids[row]], B = W ; pass 1: A = agg[row], B = M
  #pragma unroll
  for (int pass = 0; pass < 2; ++pass) {
    const float* __restrict__ X  = (pass == 0) ? (emb + (size_t)nid * DD)
                                               : (agg + (size_t)rowc * DD);
    const float* __restrict__ Bm = (pass == 0) ? W : M;
    // per-lane base into B for this lane's (column, k-half) slot
    const float* __restrict__ Bl = Bm + (size_t)rowsel * DD + kshift;

    for (int k0 = 0; k0 < DD; k0 += 4) {
      v2f a;
      a.x = X[k0 + kshift];
      a.y = X[k0 + kshift + 1];

      // Load ALL 8 B tiles first (distinct registers -> clause'd loads).
      v2f b[8];
      #pragma unroll
      for (int jt = 0; jt < 8; ++jt) {
        const float* p = Bl + (size_t)jt * 16 * DD + k0;
        b[jt].x = p[0];
        b[jt].y = p[1];
      }

      // Then issue the 8 WMMAs, draining loads with staggered waits.
      #pragma unroll
      for (int jt = 0; jt < 8; ++jt) {
        acc[jt] = __builtin_amdgcn_wmma_f32_16x16x4_f32(
            /*neg_a=*/false, a, /*neg_b=*/false, b[jt],
            /*c_mod=*/(short)0, acc[jt],
            /*reuse_a=*/false, /*reuse_b=*/false);
      }
    }
  }

  // Row-validity mask: component r of this lane corresponds to matrix row
  // node_base + r + 8*half. Zero out tail rows (only hits the last wave).
  float mask[8];
  #pragma unroll
  for (int r = 0; r < 8; ++r) {
    mask[r] = ((node_base + r + 8 * half) < N) ? 1.0f : 0.0f;
  }

  // ReLU per element, column-sum the 16x16 tiles, fold into result.
  #pragma unroll
  for (int jt = 0; jt < 8; ++jt) {
    float s = 0.0f;
    #pragma unroll
    for (int r = 0; r < 8; ++r) {
      float v = acc[jt][r];
      s += mask[r] * ((v > 0.0f) ? v : 0.0f);
    }
    s += __shfl_xor(s, 16, 32);            // combine row halves (wave32)
    if (lane < 16) {
      atomicAdd(&result[jt * 16 + rowsel], s);
    }
  }
}

// ---------------------------------------------------------------------------
// Kernel C: numerically-stable softmax over 128 values -> d_out [128,1]
// ---------------------------------------------------------------------------
__global__ void softmax128(const float* __restrict__ result,
                           float* __restrict__ out) {
  __shared__ float sdata[DD];
  const int t = threadIdx.x;
  const float v = result[t];
  sdata[t] = v;
  __syncthreads();
  for (int s = DD / 2; s > 0; s >>= 1) {
    if (t < s) sdata[t] = fmaxf(sdata[t], sdata[t + s]);
    __syncthreads();
  }
  const float mx = sdata[0];
  __syncthreads();
  const float e = expf(v - mx);
  sdata[t] = e;
  __syncthreads();
  for (int s = DD / 2; s > 0; s >>= 1) {
    if (t < s) sdata[t] += sdata[t + s];
    __syncthreads();
  }
  out[t] = e / sdata[0];
}

// ---------------------------------------------------------------------------
// Host-side launch. Input order per setup_inputs():
//   0: node_ids [N] i32, 1: neighbor_ids [E] i32, 2: segment_ids [E] i32,
//   3: W [128*128] f32, 4: M [128*128] f32, 5: emb [V*128] f32
// Output: 128 floats (softmax column vector).
// Workspace: agg [N*128] f32, then result [128] f32.
// ---------------------------------------------------------------------------
extern "C" void kernel_launch(void* const* d_in, const int* in_sizes, int n_in,
                              void* d_out, int out_size, void* d_ws, size_t ws_size,
                              hipStream_t stream) {
  const int*   node_ids     = (const int*)d_in[0];
  const int*   neighbor_ids = (const int*)d_in[1];
  const int*   segment_ids  = (const int*)d_in[2];
  const float* W            = (const float*)d_in[3];
  const float* M            = (const float*)d_in[4];
  const float* emb          = (const float*)d_in[5];

  const int N = in_sizes[0];
  const int E = in_sizes[1];

  float* agg    = (float*)d_ws;                 // N*128 floats
  float* result = agg + (size_t)N * DD;         // 128 floats
  float* out    = (float*)d_out;

  zero_result<<<1, DD, 0, stream>>>(result);

  seg_aggregate<<<N, DD, 0, stream>>>(neighbor_ids, segment_ids, emb, agg, E);

  const int wavesPerBlock = 4;                  // 4 waves * 16 nodes = 64 nodes/block
  const int nodesPerBlock = wavesPerBlock * 16;
  const int grid = (N + nodesPerBlock - 1) / nodesPerBlock;
  gemm_relu_colsum<<<grid, wavesPerBlock * 32, 0, stream>>>(
      node_ids, emb, agg, W, M, result, N);

  softmax128<<<1, DD, 0, stream>>>(result, out);
}